// _REC_BPR_LOSS_6811818132220
// MI455X (gfx1250) — compile-verified
//
#include <hip/hip_runtime.h>

// BPR pairwise ranking loss for [B, L=256] rows.
// loss = -sum over rows of sum_{valid pos p, valid neg n} log_sigmoid(pred_p - pred_n)

typedef __attribute__((ext_vector_type(2))) float v2f;
typedef __attribute__((ext_vector_type(8))) float v8f;

#define BPR_L 256
#define BPR_WAVES (BPR_L / 32)

// Sum of all 32 lane values (returned in every lane), via
// V_WMMA_F32_16X16X4_F32 against an all-ones B matrix.
// A layout (f32 16x4): lanes 0-15 hold {K=0,K=1} for M=lane, lanes 16-31 hold
// {K=2,K=3} for M=lane-16. With a=(v,0): D[m][n] = v_m + v_{m+16}.
// D layout: VGPR r, lanes 0-15 -> M=r, lanes 16-31 -> M=r+8. Summing the 8 D
// VGPRs per lane gives half the total; one xor-16 shuffle completes it.
__device__ __forceinline__ float wave_reduce_sum_wmma(float v) {
  v2f a; a.x = v;    a.y = 0.0f;
  v2f b; b.x = 1.0f; b.y = 1.0f;
  v8f c = {};
  c = __builtin_amdgcn_wmma_f32_16x16x4_f32(
      /*neg_a=*/false, a, /*neg_b=*/false, b,
      /*c_mod=*/(short)0, c, /*reuse_a=*/false, /*reuse_b=*/false);
  float s = ((c[0] + c[1]) + (c[2] + c[3])) + ((c[4] + c[5]) + (c[6] + c[7]));
  s += __shfl_xor(s, 16);
  return s;
}

// -log_sigmoid(x) = log1p(exp(-|x|)) - min(x, 0), with hardware exp/log.
// e in (0,1], so __logf(1+e) loses <= ~3e-8 absolute per term vs log1p.
__device__ __forceinline__ float neg_log_sigmoid(float x) {
  const float e = __expf(-fabsf(x));
  return __logf(1.0f + e) - fminf(x, 0.0f);
}

__global__ __launch_bounds__(BPR_L) void bpr_row_partial(
    const float* __restrict__ preds, const int* __restrict__ targets,
    const int* __restrict__ mask, float* __restrict__ partials) {
  __shared__ float s_pos[BPR_L];
  __shared__ float s_neg[BPR_L];
  __shared__ int   s_wp[BPR_WAVES];
  __shared__ int   s_wn[BPR_WAVES];
  __shared__ float s_wsum[BPR_WAVES];

  const int t    = threadIdx.x;
  const int lane = t & 31;
  const int wave = t >> 5;
  const size_t base = (size_t)blockIdx.x * BPR_L;

  const float p  = preds[base + t];
  const int   tv = targets[base + t];
  const int   mv = mask[base + t];
  const bool isp = (mv != 0) && (tv != 0);
  const bool isn = (mv != 0) && (tv == 0);

  // Deterministic stream compaction: ballot + popcount prefix (no atomics).
  const unsigned bp = (unsigned)__ballot(isp);
  const unsigned bn = (unsigned)__ballot(isn);
  if (lane == 0) { s_wp[wave] = __popc(bp); s_wn[wave] = __popc(bn); }
  __syncthreads();

  int pbase = 0, nbase = 0, np = 0, nn = 0;
#pragma unroll
  for (int w = 0; w < BPR_WAVES; ++w) {
    const int cp = s_wp[w], cn = s_wn[w];
    if (w < wave) { pbase += cp; nbase += cn; }
    np += cp; nn += cn;
  }
  const unsigned lm = (1u << lane) - 1u;
  if (isp) s_pos[pbase + __popc(bp & lm)] = p;
  if (isn) s_neg[nbase + __popc(bn & lm)] = p;
  __syncthreads();

  // 16x16 thread tile over the (positives x negatives) pair grid.
  float acc = 0.0f;
  const int tx = t & 15;
  const int ty = t >> 4;
  for (int ip = ty; ip < np; ip += 16) {
    const float pp = s_pos[ip];
#pragma unroll 4
    for (int in = tx; in < nn; in += 16) {
      acc += neg_log_sigmoid(pp - s_neg[in]);
    }
  }

  // Block reduction: WMMA wave-sum, then combine the 8 wave totals.
  const float ws = wave_reduce_sum_wmma(acc);
  if (lane == 0) s_wsum[wave] = ws;
  __syncthreads();
  if (t == 0) {
    float s = 0.0f;
#pragma unroll
    for (int w = 0; w < BPR_WAVES; ++w) s += s_wsum[w];
    partials[blockIdx.x] = s;
  }
}

__global__ __launch_bounds__(256) void bpr_reduce_final(
    const float* __restrict__ partials, int n, float* __restrict__ out) {
  __shared__ float s_wsum[8];
  const int t = threadIdx.x;
  const int lane = t & 31;
  const int wave = t >> 5;

  float acc = 0.0f;
  for (int i = t; i < n; i += 256) acc += partials[i];

  const float ws = wave_reduce_sum_wmma(acc);
  if (lane == 0) s_wsum[wave] = ws;
  __syncthreads();
  if (t == 0) {
    float s = 0.0f;
#pragma unroll
    for (int w = 0; w < 8; ++w) s += s_wsum[w];
    out[0] = s;  // partials already hold -log_sigmoid sums
  }
}

extern "C" void kernel_launch(void* const* d_in, const int* in_sizes, int n_in,
                              void* d_out, int out_size, void* d_ws, size_t ws_size,
                              hipStream_t stream) {
  (void)n_in; (void)out_size; (void)ws_size;
  const float* preds   = (const float*)d_in[0];
  const int*   targets = (const int*)d_in[1];
  const int*   mask    = (const int*)d_in[2];
  float* out      = (float*)d_out;
  float* partials = (float*)d_ws;  // B floats of scratch

  const int B = in_sizes[0] / BPR_L;

  bpr_row_partial<<<B, BPR_L, 0, stream>>>(preds, targets, mask, partials);
  bpr_reduce_final<<<1, 256, 0, stream>>>(partials, B, out);
}